// PhenotypeSimilarityLoss_15796889715192
// MI455X (gfx1250) — compile-verified
//
#include <hip/hip_runtime.h>
#include <hip/hip_bf16.h>
#include <math.h>

// PhenotypeSimilarityLoss for MI455X (gfx1250), wave32 + WMMA f32 16x16x4.
//
// B=8192, D=128. lat_sim in [-2,2] => exp never overflows => no max-shift LSE.
// Pass1: S[i] = sum_{j!=i} exp(2*dot(latn_i, latn_j))   (tiled WMMA GEMM)
// Pass2: for pos pairs: loss += -s + log(exp(s)+S[i]); abr += asim; cnt++
//
// Register blocking: each wave computes a 64x32 block (4 M-tiles x 2 N-tiles
// of 16x16). Per K-step: 6 b64 loads feed 8 WMMAs (0.75 loads/wmma vs 2.0
// for 1x1 tiles) -> inner loop is matrix-pipe-bound, not L0-load-bound.
// Working set (~4.4 MB) is fully L2-resident.

#define BN 8192
#define MB 4            // M-tiles per wave (64 rows)
#define NB 2            // N-tiles per wave (32 cols)
#define NBLK_M (BN / (16 * MB))   // 128
#define NBLK_N (BN / (16 * NB))   // 256
#define EPSN 1e-12f

typedef __attribute__((ext_vector_type(2))) float v2f;
typedef __attribute__((ext_vector_type(8))) float v8f;

struct Accum { float loss; float abr; unsigned int cnt; };

// ---------------- normalization + init ----------------

__global__ void k_norm_lat(const float* __restrict__ x, float* __restrict__ latn) {
  int row = blockIdx.x;            // 8192 blocks, 32 lanes each
  int lane = threadIdx.x;
  float4 v = ((const float4*)(x + row * 128))[lane];
  float ss = v.x*v.x + v.y*v.y + v.z*v.z + v.w*v.w;
  #pragma unroll
  for (int m = 1; m < 32; m <<= 1) ss += __shfl_xor(ss, m, 32);
  float inv = 1.0f / fmaxf(sqrtf(ss), EPSN);
  v.x *= inv; v.y *= inv; v.z *= inv; v.w *= inv;
  ((float4*)(latn + row * 128))[lane] = v;
}

__global__ void k_norm_abr(const float* __restrict__ x, float* __restrict__ abrn,
                           float* __restrict__ S, Accum* __restrict__ acc) {
  int t = blockIdx.x * blockDim.x + threadIdx.x;
  if (t < BN) {
    float v[6]; float ss = 0.0f;
    #pragma unroll
    for (int k = 0; k < 6; ++k) { v[k] = x[t * 6 + k]; ss += v[k] * v[k]; }
    float inv = 1.0f / fmaxf(sqrtf(ss), EPSN);
    #pragma unroll
    for (int k = 0; k < 6; ++k) abrn[t * 6 + k] = v[k] * inv;
    S[t] = 0.0f;
  }
  if (t == 0) { acc->loss = 0.0f; acc->abr = 0.0f; acc->cnt = 0u; }
}

// ---------------- register-blocked Gram tile (64x32 per wave) ----------------

__device__ __forceinline__ void gram_block(const float* __restrict__ latn,
                                           int bm, int bn, int lr, int h,
                                           v8f c[MB][NB]) {
  const v2f* pa[MB];
  const v2f* pb[NB];
  #pragma unroll
  for (int mt = 0; mt < MB; ++mt)
    pa[mt] = (const v2f*)(latn + ((bm * MB + mt) * 16 + lr) * 128 + h * 2);
  #pragma unroll
  for (int nt = 0; nt < NB; ++nt)
    pb[nt] = (const v2f*)(latn + ((bn * NB + nt) * 16 + lr) * 128 + h * 2);

  #pragma unroll
  for (int mt = 0; mt < MB; ++mt)
    #pragma unroll
    for (int nt = 0; nt < NB; ++nt)
      c[mt][nt] = (v8f){};

  #pragma unroll
  for (int kk = 0; kk < 32; ++kk) {   // K = 128 in steps of 4
    v2f a[MB], b[NB];
    #pragma unroll
    for (int mt = 0; mt < MB; ++mt) a[mt] = pa[mt][kk * 2];
    #pragma unroll
    for (int nt = 0; nt < NB; ++nt) b[nt] = pb[nt][kk * 2];
    #pragma unroll
    for (int mt = 0; mt < MB; ++mt)
      #pragma unroll
      for (int nt = 0; nt < NB; ++nt)
        c[mt][nt] = __builtin_amdgcn_wmma_f32_16x16x4_f32(
            false, a[mt], false, b[nt], (short)0, c[mt][nt], false, false);
  }
}

// ---------------- pass 1: row sums of exp ----------------

__global__ __launch_bounds__(256) void k_rowsum(const float* __restrict__ latn,
                                                float* __restrict__ S) {
  int wave = threadIdx.x >> 5;
  int lane = threadIdx.x & 31;
  int flat = blockIdx.x * 8 + wave;         // NBLK_M * NBLK_N waves
  int bm = flat >> 8, bn = flat & (NBLK_N - 1);
  int lr = lane & 15, h = lane >> 4;

  v8f c[MB][NB];
  gram_block(latn, bm, bn, lr, h, c);

  #pragma unroll
  for (int mt = 0; mt < MB; ++mt) {
    #pragma unroll
    for (int r = 0; r < 8; ++r) {
      int i = (bm * MB + mt) * 16 + h * 8 + r;
      float e = 0.0f;
      #pragma unroll
      for (int nt = 0; nt < NB; ++nt) {
        int j = (bn * NB + nt) * 16 + lr;
        float s = c[mt][nt][r] * 2.0f;       // / TEMPERATURE
        e += (i == j) ? 0.0f : __expf(s);    // diag excluded from LSE
      }
      // reduce over the 32 columns (16 lanes x 2 N-tiles) held by this half-wave
      #pragma unroll
      for (int m = 1; m < 16; m <<= 1) e += __shfl_xor(e, m, 32);
      if (lr == 0) atomicAdd(&S[i], e);
    }
  }
}

// ---------------- pass 2: masked accumulation ----------------

__global__ __launch_bounds__(256) void k_accum(const float* __restrict__ latn,
                                               const float* __restrict__ abrn,
                                               const float* __restrict__ S,
                                               Accum* __restrict__ acc) {
  int wave = threadIdx.x >> 5;
  int lane = threadIdx.x & 31;
  int flat = blockIdx.x * 8 + wave;
  int bm = flat >> 8, bn = flat & (NBLK_N - 1);
  int lr = lane & 15, h = lane >> 4;

  v8f c[MB][NB];
  gram_block(latn, bm, bn, lr, h, c);

  float bj[NB][6];
  #pragma unroll
  for (int nt = 0; nt < NB; ++nt) {
    int j = (bn * NB + nt) * 16 + lr;
    #pragma unroll
    for (int k = 0; k < 6; ++k) bj[nt][k] = abrn[j * 6 + k];
  }

  float lloss = 0.0f, labr = 0.0f;
  unsigned int lcnt = 0u;
  #pragma unroll
  for (int mt = 0; mt < MB; ++mt) {
    #pragma unroll
    for (int r = 0; r < 8; ++r) {
      int i = (bm * MB + mt) * 16 + h * 8 + r;
      float ai[6];
      #pragma unroll
      for (int k = 0; k < 6; ++k) ai[k] = abrn[i * 6 + k];
      float Si = S[i];                       // exp(row_lse_i)
      #pragma unroll
      for (int nt = 0; nt < NB; ++nt) {
        int j = (bn * NB + nt) * 16 + lr;
        float asim = 0.0f;
        #pragma unroll
        for (int k = 0; k < 6; ++k) asim += ai[k] * bj[nt][k];
        if ((i != j) && (asim > 0.8f)) {
          float s = c[mt][nt][r] * 2.0f;
          lloss += -s + __logf(__expf(s) + Si);
          labr  += asim;
          lcnt  += 1u;
        }
      }
    }
  }
  // full-wave reduction, then one atomic triple per wave
  #pragma unroll
  for (int m = 1; m < 32; m <<= 1) {
    lloss += __shfl_xor(lloss, m, 32);
    labr  += __shfl_xor(labr, m, 32);
    lcnt  += __shfl_xor(lcnt, m, 32);
  }
  if (lane == 0) {
    atomicAdd(&acc->loss, lloss);
    atomicAdd(&acc->abr, labr);
    atomicAdd(&acc->cnt, lcnt);
  }
}

// ---------------- finalize ----------------

__global__ void k_final(const Accum* __restrict__ acc, float* __restrict__ out) {
  if (threadIdx.x == 0 && blockIdx.x == 0) {
    unsigned int n = acc->cnt;
    float denom = (float)(n > 0u ? n : 1u);
    out[0] = (n > 0u) ? acc->loss / denom : 0.0f;   // phenotype_loss
    out[1] = (n > 0u) ? acc->abr  / denom : 0.0f;   // mean_abr_similarity
    out[2] = (float)n;                              // num_pos
    out[3] = (float)n / (float)((double)BN * (double)(BN - 1)); // pos ratio
  }
}

extern "C" void kernel_launch(void* const* d_in, const int* in_sizes, int n_in,
                              void* d_out, int out_size, void* d_ws, size_t ws_size,
                              hipStream_t stream) {
  const float* lat = (const float*)d_in[0];   // [8192,128] f32
  const float* abr = (const float*)d_in[1];   // [8192,6] f32

  // Workspace layout (~4.43 MB): latn | abrn | S | accum
  float* ws   = (float*)d_ws;
  float* latn = ws;                               // 8192*128
  float* abrn = latn + BN * 128;                  // 8192*6
  float* S    = abrn + BN * 6;                    // 8192
  Accum* acc  = (Accum*)(S + BN);

  int nwaves = NBLK_M * NBLK_N;                   // 32768 waves per pass
  k_norm_lat<<<BN, 32, 0, stream>>>(lat, latn);
  k_norm_abr<<<(BN + 255) / 256, 256, 0, stream>>>(abr, abrn, S, acc);
  k_rowsum<<<nwaves / 8, 256, 0, stream>>>(latn, S);
  k_accum<<<nwaves / 8, 256, 0, stream>>>(latn, abrn, S, acc);
  k_final<<<1, 32, 0, stream>>>(acc, (float*)d_out);
}